// LightweightConv1d_36395552867057
// MI455X (gfx1250) — compile-verified
//
#include <hip/hip_runtime.h>

#define TLEN 4096      // time dimension
#define KTAPS 7
#define CPAD 3
#define THREADS 256
#define PER_THREAD (TLEN / THREADS)   // 16 outputs per thread
#define HALO 4                        // 16B-aligned halo on each side

typedef float v4f __attribute__((ext_vector_type(4)));   // native vector -> b128 ops

__global__ __launch_bounds__(THREADS)
void lconv1d_kernel(const float* __restrict__ x,
                    const float* __restrict__ weight,
                    float* __restrict__ out,
                    int head_mask)        // heads - 1 (heads is a power of two: 16)
{
    __shared__ __align__(16) float tile[HALO + TLEN + HALO];

    const int r   = blockIdx.x;          // row in [0, B*C)
    const int tid = threadIdx.x;
    const float* xrow = x   + (size_t)r * TLEN;
    float*       orow = out + (size_t)r * TLEN;

    // Zero the halos (disjoint from the async-written region).
    if (tid < HALO) {
        tile[tid] = 0.0f;
        tile[HALO + TLEN + tid] = 0.0f;
    }

    // ---- Stage the row into LDS via CDNA5 async global->LDS (ASYNCcnt path).
    // Non-temporal: every input byte is consumed exactly once (reuse lives in
    // LDS), so don't let 256 MB of dead lines churn the 192 MB L2.
#pragma unroll
    for (int p = 0; p < TLEN / (THREADS * 4); ++p) {
        const int i = p * THREADS * 4 + tid * 4;
        // Low 32 bits of a generic LDS pointer == wave-relative LDS byte offset
        // (ISA 10.2: LDS aperture; the low 32 address bits are the LDS address).
        unsigned lds_addr  = (unsigned)(size_t)(&tile[HALO + i]);
        const float* gaddr = xrow + i;
        asm volatile("global_load_async_to_lds_b128 %0, %1, off th:TH_LOAD_NT"
                     :: "v"(lds_addr), "v"(gaddr)
                     : "memory");
    }
    asm volatile("s_wait_asynccnt 0" ::: "memory");
    __syncthreads();

    // ---- Softmax over the 7 taps (uniform per block -> SALU/SMEM path) ----
    const int h = r & head_mask;
    float wv[KTAPS];
    float m = -__builtin_inff();
#pragma unroll
    for (int k = 0; k < KTAPS; ++k) {
        wv[k] = weight[h * KTAPS + k];
        m = fmaxf(m, wv[k]);
    }
    float s = 0.0f;
#pragma unroll
    for (int k = 0; k < KTAPS; ++k) {
        wv[k] = __expf(wv[k] - m);
        s += wv[k];
    }
    const float inv = 1.0f / s;
#pragma unroll
    for (int k = 0; k < KTAPS; ++k) wv[k] *= inv;

    // ---- 16 outputs per thread from an LDS register window ----
    const int t0 = tid * PER_THREAD;

    float win[24];   // inputs tile[t0 .. t0+23] (need [t0+1 .. t0+22])
#pragma unroll
    for (int j = 0; j < 6; ++j) {
        *(v4f*)&win[j * 4] = *(const v4f*)&tile[t0 + j * 4];   // ds_load_b128
    }

    float acc[PER_THREAD];
#pragma unroll
    for (int t = 0; t < PER_THREAD; ++t) {
        // input tile index = HALO + (t0 + t) + k - CPAD = t0 + (t + k + 1)
        float a = 0.0f;
#pragma unroll
        for (int k = 0; k < KTAPS; ++k) {
            a = fmaf(wv[k], win[t + k + 1], a);
        }
        acc[t] = a;
    }

    // Write-once output: non-temporal b128 stores (th:TH_STORE_NT).
#pragma unroll
    for (int t = 0; t < PER_THREAD; t += 4) {
        v4f v = *(const v4f*)&acc[t];
        __builtin_nontemporal_store(v, (v4f*)&orow[t0 + t]);
    }
}

extern "C" void kernel_launch(void* const* d_in, const int* in_sizes, int n_in,
                              void* d_out, int out_size, void* d_ws, size_t ws_size,
                              hipStream_t stream)
{
    const float* x = (const float*)d_in[0];   // (16, 1024, 4096) fp32
    const float* w = (const float*)d_in[1];   // (16, 1, 7) fp32
    float* out = (float*)d_out;

    const int rows  = in_sizes[0] / TLEN;     // B*C = 16384
    const int heads = in_sizes[1] / KTAPS;    // 16 (power of two)

    lconv1d_kernel<<<dim3(rows), dim3(THREADS), 0, stream>>>(x, w, out, heads - 1);
}